// LocalAttention_52750788329729
// MI455X (gfx1250) — compile-verified
//
#include <hip/hip_runtime.h>
#include <math.h>

// LocalAttention on MI455X (gfx1250, wave32).
// Memory-bound (~2 FLOP/byte): single-pass online-softmax streaming of x
// (256 MB -> ~11us at 23.3 TB/s). Subtiles are DMA'd straight into LDS with
// GLOBAL_LOAD_ASYNC_TO_LDS_B128 (ASYNCcnt), double-buffered so load(n+1)
// overlaps compute(n). WMMA f32 16x16x4 crunches the window-3 conv dot
// products from LDS; its 16x N-waste is free under the HBM roof.

#define B_SZ    32
#define L_SZ    4096
#define D_SZ    512
#define WINSZ   3
#define CHUNK   256
#define NCHUNK  (L_SZ / CHUNK)   // 16 -> 512 workgroups in kernel 1
#define TILE    16
#define NSUB    (CHUNK / TILE)   // 16 subtiles per chunk
#define PITCH   516              // LDS row pitch (floats); 516%64=4 kills bank conflicts
#define THREADS 256
#define REC     (D_SZ + 2)       // per-(b,chunk) record: m, s, acc[512]

typedef float v2f __attribute__((ext_vector_type(2)));
typedef float v8f __attribute__((ext_vector_type(8)));

// Issue one 18-row halo tile (rows t0-1 .. t0+16) as async DMA into LDS.
// In-range lanes: global_load_async_to_lds_b128 (memory -> LDS, ASYNCcnt).
// Out-of-range halo lanes (only at l=0 / l=L-1): plain zero store to LDS.
__device__ __forceinline__ void issue_tile_load(const float* __restrict__ xg,
                                                float* lds_tile, int t0, int tid) {
#pragma unroll
    for (int i = 0; i < 9; ++i) {
        int idx = tid + THREADS * i;       // 0..2303 = 18 rows * 128 float4
        int row = idx >> 7;
        int c4  = idx & 127;
        int g   = t0 - 1 + row;
        float* dst = &lds_tile[row * PITCH + (c4 << 2)];
        if (g >= 0 && g < L_SZ) {
            unsigned int       ldsoff = (unsigned int)(size_t)dst;   // addr[31:0] = LDS offset
            unsigned long long gaddr  = (unsigned long long)(xg + (size_t)g * D_SZ + (c4 << 2));
            asm volatile("global_load_async_to_lds_b128 %0, %1, off"
                         :: "v"(ldsoff), "v"(gaddr) : "memory");
        } else {
            *(float4*)dst = make_float4(0.0f, 0.0f, 0.0f, 0.0f);
        }
    }
}

__global__ __launch_bounds__(THREADS)
void la_partial_kernel(const float* __restrict__ x,
                       const float* __restrict__ w,
                       const float* __restrict__ bias,
                       float* __restrict__ ws) {
    __shared__ float xt0[(TILE + 2) * PITCH];  // ping (~36 KB)
    __shared__ float xt1[(TILE + 2) * PITCH];  // pong (~36 KB)
    __shared__ float w_lds[WINSZ * D_SZ];      // 6 KB
    __shared__ float y_lds[2][TILE];

    const int chunk = blockIdx.x;
    const int b     = blockIdx.y;
    const int tid   = threadIdx.x;
    const int lane  = tid & 31;
    const int wave  = tid >> 5;

    const float* xg = x + (size_t)b * L_SZ * D_SZ;
    for (int i = tid; i < WINSZ * D_SZ; i += THREADS) w_lds[i] = w[i];
    if (tid < TILE) y_lds[0][tid] = 0.0f;

    const float bv = bias[0];
    float  run_m = -INFINITY;
    float  run_s = 0.0f;
    float2 acc   = make_float2(0.0f, 0.0f);
    const int dcol = tid * 2;                  // this thread owns d = dcol, dcol+1

    const int l0chunk = chunk * CHUNK;
    issue_tile_load(xg, xt0, l0chunk, tid);    // prime the pipeline

    for (int sub = 0; sub < NSUB; ++sub) {
        float* cur   = (sub & 1) ? xt1 : xt0;
        float* nxt   = (sub & 1) ? xt0 : xt1;
        float* y_cur = y_lds[sub & 1];
        float* y_nxt = y_lds[(sub & 1) ^ 1];
        const int t0 = l0chunk + sub * TILE;

        // current tile's DMA complete; also orders w_lds staging / y resets
        asm volatile("s_wait_asynccnt 0x0" ::: "memory");
        __syncthreads();

        // overlap: DMA subtile n+1 into the other buffer while we compute
        if (sub + 1 < NSUB) issue_tile_load(xg, nxt, t0 + TILE, tid);

        // ---- conv via V_WMMA_F32_16X16X4_F32 ----
        // A[m,k] = cur[m+j][d0+k] (window row), B[k,n] = w[j][d0+k] broadcast
        // over all 16 columns n, so C[m,n] == partial y[t0+m] for every n.
        // Each of the 8 waves covers a 64-wide d-slice for all 3 taps.
        v8f c = {};
        const int m  = lane & 15;              // A row (ISA 7.12.2: 16x4 f32 A)
        const int kk = (lane < 16) ? 0 : 2;    // lanes 0-15: K 0/1, 16-31: K 2/3
        const int dbase = wave * 64;
        for (int j = 0; j < WINSZ; ++j) {
            const float* xr = &cur[(m + j) * PITCH];
            const float* wj = &w_lds[j * D_SZ];
            for (int i = 0; i < 16; ++i) {
                int d0 = dbase + (i << 2) + kk;
                v2f a;  a.x  = xr[d0]; a.y  = xr[d0 + 1];
                v2f bm; bm.x = wj[d0]; bm.y = wj[d0 + 1];
                c = __builtin_amdgcn_wmma_f32_16x16x4_f32(
                        false, a, false, bm, (short)0, c, false, false);
            }
        }
        // C layout: VGPR r -> (M=r, lanes 0-15) / (M=8+r, lanes 16-31); read N=0.
        if (lane == 0) {
            for (int r = 0; r < 8; ++r) atomicAdd(&y_cur[r], c[r]);
        } else if (lane == 16) {
            for (int r = 0; r < 8; ++r) atomicAdd(&y_cur[8 + r], c[r]);
        }
        __syncthreads();

        // ---- online softmax update (redundant per thread => consistent) ----
        float yv[TILE];
        float tmax = -INFINITY;
        for (int l = 0; l < TILE; ++l) {
            yv[l] = y_cur[l] + bv;
            tmax  = fmaxf(tmax, yv[l]);
        }
        const float m_new = fmaxf(run_m, tmax);
        const float scale = __expf(run_m - m_new);   // 0 on first subtile
        float e[TILE], ssum = 0.0f;
        for (int l = 0; l < TILE; ++l) {
            e[l] = __expf(yv[l] - m_new);
            ssum += e[l];
        }
        run_s = run_s * scale + ssum;
        acc.x *= scale; acc.y *= scale;
        for (int l = 0; l < TILE; ++l) {
            const float* xr = &cur[(l + 1) * PITCH]; // row t0+l lives at index l+1
            acc.x = fmaf(e[l], xr[dcol],     acc.x);
            acc.y = fmaf(e[l], xr[dcol + 1], acc.y);
        }
        run_m = m_new;

        // reset the *next* y buffer; next iteration's top barrier orders this
        // against the next conv's atomic adds
        if (tid < TILE) y_nxt[tid] = 0.0f;
    }

    float* rec = ws + (size_t)(b * NCHUNK + chunk) * REC;
    if (tid == 0) { rec[0] = run_m; rec[1] = run_s; }
    rec[2 + dcol]     = acc.x;
    rec[2 + dcol + 1] = acc.y;
}

__global__ __launch_bounds__(THREADS)
void la_combine_kernel(const float* __restrict__ ws, float* __restrict__ out) {
    const int b   = blockIdx.x;
    const int tid = threadIdx.x;
    const float* base = ws + (size_t)b * NCHUNK * REC;

    float M = -INFINITY;
    for (int c = 0; c < NCHUNK; ++c) M = fmaxf(M, base[c * REC]);
    float S = 0.0f;
    for (int c = 0; c < NCHUNK; ++c) S += base[c * REC + 1] * __expf(base[c * REC] - M);
    const float inv = 1.0f / S;

    const int d = tid * 2;
    float2 o = make_float2(0.0f, 0.0f);
    for (int c = 0; c < NCHUNK; ++c) {
        const float f = __expf(base[c * REC] - M);
        o.x = fmaf(f, base[c * REC + 2 + d],     o.x);
        o.y = fmaf(f, base[c * REC + 2 + d + 1], o.y);
    }
    out[b * D_SZ + d]     = o.x * inv;
    out[b * D_SZ + d + 1] = o.y * inv;
}

extern "C" void kernel_launch(void* const* d_in, const int* in_sizes, int n_in,
                              void* d_out, int out_size, void* d_ws, size_t ws_size,
                              hipStream_t stream) {
    const float* x    = (const float*)d_in[0];   // [32,4096,512] f32
    const float* w    = (const float*)d_in[1];   // [3,512] f32
    const float* bias = (const float*)d_in[2];   // scalar f32
    float* out = (float*)d_out;                  // [32,512] f32
    float* ws  = (float*)d_ws;                   // needs 32*16*514*4 ~= 1.05 MB

    dim3 grid1(NCHUNK, B_SZ);
    la_partial_kernel<<<grid1, THREADS, 0, stream>>>(x, w, bias, ws);
    la_combine_kernel<<<B_SZ, THREADS, 0, stream>>>(ws, out);
    (void)in_sizes; (void)n_in; (void)out_size; (void)ws_size;
}